// Loss_13975823581336
// MI455X (gfx1250) — compile-verified
//
#include <hip/hip_runtime.h>
#include <hip/hip_bf16.h>

#define BB 64
#define NN 8732
#define CC 81
#define CHUNKS 69               // ceil(NN/128)
#define NPAD 9216               // ceil(NN/1024)*1024

typedef float v2f __attribute__((ext_vector_type(2)));
typedef float v8f __attribute__((ext_vector_type(8)));

// ---------------- Kernel 1: per-anchor CE (logsumexp via WMMA) + smooth-L1 partials ----
__global__ __launch_bounds__(256) void k1_con_loc(
    const float* __restrict__ ploc, const float* __restrict__ plabel,
    const float* __restrict__ gloc, const int* __restrict__ glabel,
    const float* __restrict__ dboxes,
    float* __restrict__ ws_con, float* __restrict__ ws_loc, unsigned* __restrict__ ws_pos)
{
    const int b   = blockIdx.y;
    const int tid = threadIdx.x;
    const int w   = tid >> 5;
    const int l   = tid & 31;
    const int h   = l >> 4;                 // half: 0 -> residues {0,1}, 1 -> {2,3}
    const int anchor = blockIdx.x * 128 + w * 16 + (l & 15);
    const bool valid = anchor < NN;
    const int a   = valid ? anchor : (NN - 1);
    const int p   = h * 2;

    const float* pl = plabel + (size_t)b * CC * NN;
    const int gl = glabel[(size_t)b * NN + a];

    // Pass A: load 40 classes (2 residues x 20 chunks) into registers, track max + label logit
    float xs[40];
    float m = -3.4e38f, xl = -3.4e38f;
#pragma unroll
    for (int j = 0; j < 20; ++j) {
        const int c0 = 4 * j + p;
        float x0 = pl[(size_t)c0 * NN + a];
        float x1 = pl[(size_t)(c0 + 1) * NN + a];
        xs[2 * j]     = x0;
        xs[2 * j + 1] = x1;
        m  = fmaxf(m, fmaxf(x0, x1));
        xl = (c0 == gl)     ? x0 : xl;
        xl = (c0 + 1 == gl) ? x1 : xl;
    }
    float x80 = pl[(size_t)80 * NN + a];
    m  = fmaxf(m, x80);
    xl = (gl == 80) ? x80 : xl;
    // combine the two half-wave partials (each half saw 40 of 81 classes)
    m  = fmaxf(m,  __shfl_xor(m,  16, 32));
    xl = fmaxf(xl, __shfl_xor(xl, 16, 32));

    // Pass B: sum of exp via V_WMMA_F32_16X16X4_F32 with all-ones A (exact f32 column sums)
    // Args are max-shifted (<= 0), so hardware __expf (v_exp_f32) is safe and ~1 ulp.
    v2f aones; aones.x = 1.0f; aones.y = 1.0f;
    v8f acc = {};
#pragma unroll
    for (int j = 0; j < 20; ++j) {
        v2f bb;
        bb.x = __expf(xs[2 * j]     - m);
        bb.y = __expf(xs[2 * j + 1] - m);
        acc = __builtin_amdgcn_wmma_f32_16x16x4_f32(
            false, aones, false, bb, (short)0, acc, false, false);
    }
    float s   = acc[0] + __expf(x80 - m);   // D row is identical for all M; lane n reads col n
    float con = m + __logf(s) - xl;

    // Smooth-L1 location loss (masked) computed by the low half only
    float locv = 0.f;
    unsigned posv = 0u;
    if (h == 0 && valid) {
        ws_con[(size_t)b * NN + anchor] = con;
        const bool pos = gl > 0;
        posv = pos ? 1u : 0u;
        if (pos) {
            const float* pb = ploc + (size_t)b * 4 * NN;
            const float* gb = gloc + (size_t)b * 4 * NN;
            float d0 = dboxes[0 * NN + a], d1 = dboxes[1 * NN + a];
            float d2 = dboxes[2 * NN + a], d3 = dboxes[3 * NN + a];
            float v0 = 10.f * (gb[0 * NN + a] - d0) / d2;
            float v1 = 10.f * (gb[1 * NN + a] - d1) / d3;
            float v2 = 5.f * __logf(gb[2 * NN + a] / d2);
            float v3 = 5.f * __logf(gb[3 * NN + a] / d3);
            float q, ad, sl = 0.f;
            q = pb[0 * NN + a] - v0; ad = fabsf(q); sl += (ad < 1.f) ? 0.5f * q * q : ad - 0.5f;
            q = pb[1 * NN + a] - v1; ad = fabsf(q); sl += (ad < 1.f) ? 0.5f * q * q : ad - 0.5f;
            q = pb[2 * NN + a] - v2; ad = fabsf(q); sl += (ad < 1.f) ? 0.5f * q * q : ad - 0.5f;
            q = pb[3 * NN + a] - v3; ad = fabsf(q); sl += (ad < 1.f) ? 0.5f * q * q : ad - 0.5f;
            locv = sl;
        }
    }

    // deterministic block reduction (wave shfl + LDS), no atomics
    for (int o = 16; o; o >>= 1) {
        locv += __shfl_down(locv, o, 32);
        posv += __shfl_down(posv, o, 32);
    }
    __shared__ float    s_l[8];
    __shared__ unsigned s_p[8];
    if (l == 0) { s_l[w] = locv; s_p[w] = posv; }
    __syncthreads();
    if (tid == 0) {
        float lsum = 0.f; unsigned psum = 0u;
        for (int q = 0; q < 8; ++q) { lsum += s_l[q]; psum += s_p[q]; }
        ws_loc[(size_t)b * CHUNKS + blockIdx.x] = lsum;
        ws_pos[(size_t)b * CHUNKS + blockIdx.x] = psum;
    }
}

// ---------------- Kernel 2: per-row radix select (top-k hard negatives) + final sums ----
__global__ __launch_bounds__(1024) void k2_select(
    const int* __restrict__ glabel,
    const float* __restrict__ ws_con, const float* __restrict__ ws_loc,
    const unsigned* __restrict__ ws_pos, float* __restrict__ ws_row)
{
    const int b   = blockIdx.x;
    const int tid = threadIdx.x;
    const int w   = tid >> 5;
    const int ln  = tid & 31;

    __shared__ float    s_con[NN];                // 34928 B
    __shared__ unsigned s_pmask[(NN + 31) / 32];  // 273 words
    __shared__ unsigned s_hist[256];
    __shared__ unsigned s_wcnt[32];
    __shared__ float    s_wred[32];
    __shared__ unsigned s_sel[2];
    __shared__ unsigned s_base;
    __shared__ float    s_locloss;
    __shared__ int      s_posnum;

    const size_t bN = (size_t)b * NN;

    for (int i = tid; i < NN; i += 1024) s_con[i] = ws_con[bN + i];
    for (int i0 = 0; i0 < NPAD; i0 += 1024) {
        int i = i0 + tid;
        int gl = (i < NN) ? glabel[bN + i] : 0;
        bool pos = (i < NN) && (gl > 0);
        unsigned bal = (unsigned)__ballot(pos);
        if (ln == 0 && i < NN) s_pmask[i >> 5] = bal;
    }
    if (tid == 0) {
        float lacc = 0.f; unsigned pacc = 0u;
        for (int q = 0; q < CHUNKS; ++q) {
            lacc += ws_loc[(size_t)b * CHUNKS + q];
            pacc += ws_pos[(size_t)b * CHUNKS + q];
        }
        s_locloss = lacc; s_posnum = (int)pacc; s_base = 0u;
    }
    __syncthreads();

    const int pos_num = s_posnum;
    if (pos_num == 0) { if (tid == 0) ws_row[b] = 0.f; return; }
    const unsigned k = min(3u * (unsigned)pos_num, (unsigned)NN);

    // key = float bits of con_neg (all >= 0 -> uint order == float order)
    auto keyAt = [&](int i) -> unsigned {
        bool pi = (s_pmask[i >> 5] >> (i & 31)) & 1u;
        return pi ? 0u : __float_as_uint(fmaxf(s_con[i], 0.f));
    };

    // MSD radix select: find k-th largest key and its tie rank
    unsigned pref = 0u, r = k;
    for (int pass = 0; pass < 4; ++pass) {
        const int shift = 24 - 8 * pass;
        if (tid < 256) s_hist[tid] = 0u;
        __syncthreads();
        for (int i = tid; i < NN; i += 1024) {
            unsigned u = keyAt(i);
            bool match = (pass == 0) || ((u >> (shift + 8)) == (pref >> (shift + 8)));
            if (match) atomicAdd(&s_hist[(u >> shift) & 255u], 1u);
        }
        __syncthreads();
        if (tid < 32) {
            const int dhi = 255 - 8 * tid;
            unsigned csum = 0u;
            for (int d = dhi; d > dhi - 8; --d) csum += s_hist[d];
            unsigned incl = csum;
            for (int o = 1; o < 32; o <<= 1) {
                unsigned v = __shfl_up(incl, o, 32);
                if (tid >= o) incl += v;
            }
            unsigned excl = incl - csum;
            if (excl < r && incl >= r) {           // unique crossing lane
                unsigned cum = excl;
                for (int d = dhi; d > dhi - 8; --d) {
                    cum += s_hist[d];
                    if (cum >= r) {
                        s_sel[0] = (unsigned)d;
                        s_sel[1] = r - (cum - s_hist[d]);
                        break;
                    }
                }
            }
        }
        __syncthreads();
        pref |= s_sel[0] << shift;
        r = s_sel[1];
    }
    const unsigned T = pref;          // k-th largest key
    const unsigned m_ties = r;        // how many == T to take, in index order

    // con_loss = sum(con*mask) + sum(con | key>T) + first m_ties of (key==T) in index order
    float part = 0.f;
    for (int i0 = 0; i0 < NPAD; i0 += 1024) {
        int i = i0 + tid;
        bool inb = i < NN;
        float c  = inb ? s_con[i] : 0.f;
        bool pi  = inb && ((s_pmask[i >> 5] >> (i & 31)) & 1u);
        unsigned u = (inb && !pi) ? __float_as_uint(fmaxf(c, 0.f)) : 0u;
        if (pi) part += c;
        if (inb && !pi && u > T) part += c;
        bool tie = inb && ((pi ? 0u : u) == T);
        unsigned bal = (unsigned)__ballot(tie);
        if (ln == 0) s_wcnt[w] = (unsigned)__popc(bal);
        __syncthreads();
        unsigned woff = 0u;
        for (int q = 0; q < w; ++q) woff += s_wcnt[q];
        unsigned lpre = (unsigned)__popc(bal & ((1u << ln) - 1u));
        unsigned gidx = s_base + woff + lpre;
        if (tie && gidx < m_ties) part += c;
        __syncthreads();
        if (tid == 0) {
            unsigned tot = 0u;
            for (int q = 0; q < 32; ++q) tot += s_wcnt[q];
            s_base += tot;
        }
        __syncthreads();
    }

    for (int o = 16; o; o >>= 1) part += __shfl_down(part, o, 32);
    if (ln == 0) s_wred[w] = part;
    __syncthreads();
    if (tid == 0) {
        float tot = 0.f;
        for (int q = 0; q < 32; ++q) tot += s_wred[q];
        ws_row[b] = (s_locloss + tot) / fmaxf((float)pos_num, 1e-6f);
    }
}

// ---------------- Kernel 3: mean over batch ----
__global__ void k3_mean(const float* __restrict__ ws_row, float* __restrict__ out)
{
    if (threadIdx.x == 0) {
        float s = 0.f;
        for (int b = 0; b < BB; ++b) s += ws_row[b];
        out[0] = s / (float)BB;
    }
}

extern "C" void kernel_launch(void* const* d_in, const int* in_sizes, int n_in,
                              void* d_out, int out_size, void* d_ws, size_t ws_size,
                              hipStream_t stream) {
    (void)in_sizes; (void)n_in; (void)out_size; (void)ws_size;
    const float* ploc   = (const float*)d_in[0];
    const float* plabel = (const float*)d_in[1];
    const float* gloc   = (const float*)d_in[2];
    const int*   glabel = (const int*)d_in[3];
    const float* dboxes = (const float*)d_in[4];
    float* out = (float*)d_out;

    float*    ws_con = (float*)d_ws;                         // BB*NN floats
    float*    ws_loc = ws_con + (size_t)BB * NN;             // BB*CHUNKS floats
    unsigned* ws_pos = (unsigned*)(ws_loc + (size_t)BB * CHUNKS);
    float*    ws_row = (float*)(ws_pos + (size_t)BB * CHUNKS);

    dim3 g1(CHUNKS, BB);
    k1_con_loc<<<g1, 256, 0, stream>>>(ploc, plabel, gloc, glabel, dboxes,
                                       ws_con, ws_loc, ws_pos);
    k2_select<<<BB, 1024, 0, stream>>>(glabel, ws_con, ws_loc, ws_pos, ws_row);
    k3_mean<<<1, 32, 0, stream>>>(ws_row, out);
}